// SingleChainBaseline_34626026340835
// MI455X (gfx1250) — compile-verified
//
#include <hip/hip_runtime.h>

// GCN: 3x (x@W -> dinv-scale -> edge scatter-add -> dinv*acc + b -> relu) + MLP head.
// Memory-bound on edge gather/scatter (~5 GB); GEMMs use fp32 WMMA (v_wmma_f32_16x16x4_f32).

#define NN 50000
#define NE 800000

typedef float v2f __attribute__((ext_vector_type(2)));
typedef float v8f __attribute__((ext_vector_type(8)));

// ---------------- degree / normalization ----------------
__global__ void k_init_deg(float* deg, int n) {
  int i = blockIdx.x * blockDim.x + threadIdx.x;
  if (i < n) deg[i] = 1.0f;  // self-loop
}

__global__ void k_accum_deg(float* deg, const int* __restrict__ dst, int e) {
  int i = blockIdx.x * blockDim.x + threadIdx.x;
  if (i < e)
    __hip_atomic_fetch_add(&deg[dst[i]], 1.0f, __ATOMIC_RELAXED, __HIP_MEMORY_SCOPE_AGENT);
}

__global__ void k_rsqrt_inplace(float* deg, int n) {
  int i = blockIdx.x * blockDim.x + threadIdx.x;
  if (i < n) deg[i] = rsqrtf(deg[i]);
}

// ---------------- WMMA GEMM: out = x[N,128] @ W[128,C] ----------------
// MODE 0: hs = dinv[row] * gemm; out1=hs, out2=hs (acc seeded with self-loop term)
// MODE 1: out1 = relu(gemm + bias[col])
template <int MODE, int C>
__global__ __launch_bounds__(128) void k_gemm_wmma(
    const float* __restrict__ x, const float* __restrict__ W,
    const float* __restrict__ bias, const float* __restrict__ dinv,
    float* __restrict__ out1, float* __restrict__ out2) {
  __shared__ float lds[16 * 132];  // 16 rows x 128, +4 pad to avoid bank conflicts
  const int tile = blockIdx.x;     // 16-row tile; N=50000 = 3125*16 exactly
  const int tid = threadIdx.x;

  // Stage A strip (16x128 f32) into LDS, float4 coalesced
  {
    int row = tid >> 3;
    int colB = (tid & 7) * 16;
    const float* src = x + ((size_t)(tile * 16 + row)) * 128 + colB;
    float* dstp = lds + row * 132 + colB;
#pragma unroll
    for (int j = 0; j < 4; ++j)
      *reinterpret_cast<float4*>(dstp + j * 4) =
          *reinterpret_cast<const float4*>(src + j * 4);
  }
  __syncthreads();

  const int wave = tid >> 5;
  const int lane = tid & 31;
  const int ln = lane & 15;
  const int hi = lane >> 4;      // 0 or 1
  const int khalf = hi * 2;      // lanes 16-31 hold K=k+2,k+3 (ISA 7.12.2)
  const float* arow = lds + ln * 132 + khalf;
  const int nTiles = C / 16;

  for (int t = wave; t < nTiles; t += 4) {
    const int n0 = t * 16;
    v8f acc = {0.f, 0.f, 0.f, 0.f, 0.f, 0.f, 0.f, 0.f};
    const float* wp = W + khalf * C + n0 + ln;
#pragma unroll 4
    for (int k = 0; k < 128; k += 4) {
      v2f a = *reinterpret_cast<const v2f*>(arow + k);  // ds_load_b64
      v2f b;
      b.x = wp[k * C];
      b.y = wp[(k + 1) * C];
      acc = __builtin_amdgcn_wmma_f32_16x16x4_f32(false, a, false, b,
                                                  (short)0, acc, false, false);
    }
    // D layout: VGPR v -> row v (lanes 0-15) / row v+8 (lanes 16-31), col = ln
    const int mbase = hi * 8;
    const int col = n0 + ln;
#pragma unroll
    for (int v = 0; v < 8; ++v) {
      int gr = tile * 16 + mbase + v;
      float val = acc[v];
      if (MODE == 0) {
        float h = dinv[gr] * val;
        out1[(size_t)gr * C + col] = h;
        out2[(size_t)gr * C + col] = h;  // acc init = self-loop contribution
      } else {
        val += bias[col];
        val = fmaxf(val, 0.0f);
        out1[(size_t)gr * C + col] = val;
      }
    }
  }
}

// ---------------- edge scatter: acc[dst] += hs[src] ----------------
__global__ void k_scatter(const float* __restrict__ hs, float* __restrict__ acc,
                          const int* __restrict__ src, const int* __restrict__ dst,
                          int e) {
  long long tid = (long long)blockIdx.x * blockDim.x + threadIdx.x;
  int edge = (int)(tid >> 7);
  int c = (int)(tid & 127);
  if (edge < e) {
    int s = src[edge];
    int d = dst[edge];
    __hip_atomic_fetch_add(&acc[(size_t)d * 128 + c], hs[(size_t)s * 128 + c],
                           __ATOMIC_RELAXED, __HIP_MEMORY_SCOPE_AGENT);
  }
}

// ---------------- per-layer epilogue: x' = [relu](dinv*acc + b) ----------------
template <bool RELU>
__global__ void k_epilogue(const float* __restrict__ acc,
                           const float* __restrict__ dinv,
                           const float* __restrict__ bias,
                           float* __restrict__ xout, int n) {
  int tid = blockIdx.x * blockDim.x + threadIdx.x;
  int i = tid >> 7;
  int c = tid & 127;
  if (i < n) {
    float v = dinv[i] * acc[(size_t)i * 128 + c] + bias[c];
    if (RELU) v = fmaxf(v, 0.0f);
    xout[(size_t)i * 128 + c] = v;
  }
}

// ---------------- classifier head 2: out[N,8] = h[N,64] @ cW2 + cb2 ----------------
__global__ void k_cls2(const float* __restrict__ h, const float* __restrict__ W,
                       const float* __restrict__ b, float* __restrict__ out, int n) {
  int tid = blockIdx.x * blockDim.x + threadIdx.x;
  int i = tid >> 3;
  int o = tid & 7;
  if (i < n) {
    float s = b[o];
    const float* hr = h + (size_t)i * 64;
#pragma unroll 8
    for (int k = 0; k < 64; ++k) s += hr[k] * W[k * 8 + o];
    out[(size_t)i * 8 + o] = s;
  }
}

extern "C" void kernel_launch(void* const* d_in, const int* in_sizes, int n_in,
                              void* d_out, int out_size, void* d_ws, size_t ws_size,
                              hipStream_t stream) {
  (void)in_sizes; (void)n_in; (void)out_size; (void)ws_size;
  const float* xin = (const float*)d_in[0];
  const int* ei = (const int*)d_in[1];  // [2,E] flat: row0=src, row1=dst
  const float* W0 = (const float*)d_in[2];
  const float* b0 = (const float*)d_in[3];
  const float* W1 = (const float*)d_in[4];
  const float* b1 = (const float*)d_in[5];
  const float* W2 = (const float*)d_in[6];
  const float* b2 = (const float*)d_in[7];
  const float* cW1 = (const float*)d_in[8];
  const float* cb1 = (const float*)d_in[9];
  const float* cW2 = (const float*)d_in[10];
  const float* cb2 = (const float*)d_in[11];
  float* out = (float*)d_out;

  const int n = NN, e = NE;
  const int* srcIdx = ei;
  const int* dstIdx = ei + e;

  // workspace carve-out: dinv[N] | B0[N*128] | B1[N*128] | ACC[N*128]  (~77 MB)
  char* ws = (char*)d_ws;
  float* dinv = (float*)ws;
  ws += (((size_t)n * sizeof(float)) + 255) & ~(size_t)255;
  float* B0 = (float*)ws;  ws += (size_t)n * 128 * sizeof(float);
  float* B1 = (float*)ws;  ws += (size_t)n * 128 * sizeof(float);
  float* ACC = (float*)ws;

  dim3 blk(256);
  k_init_deg<<<(n + 255) / 256, blk, 0, stream>>>(dinv, n);
  k_accum_deg<<<(e + 255) / 256, blk, 0, stream>>>(dinv, dstIdx, e);
  k_rsqrt_inplace<<<(n + 255) / 256, blk, 0, stream>>>(dinv, n);

  const int rowTiles = n / 16;  // 3125
  dim3 gblk(128);
  int sBlocks = (int)(((long long)e * 128 + 255) / 256);
  int eBlocks = (n * 128 + 255) / 256;

  // layer 0
  k_gemm_wmma<0, 128><<<rowTiles, gblk, 0, stream>>>(xin, W0, nullptr, dinv, B0, ACC);
  k_scatter<<<sBlocks, blk, 0, stream>>>(B0, ACC, srcIdx, dstIdx, e);
  k_epilogue<true><<<eBlocks, blk, 0, stream>>>(ACC, dinv, b0, B1, n);
  // layer 1
  k_gemm_wmma<0, 128><<<rowTiles, gblk, 0, stream>>>(B1, W1, nullptr, dinv, B0, ACC);
  k_scatter<<<sBlocks, blk, 0, stream>>>(B0, ACC, srcIdx, dstIdx, e);
  k_epilogue<true><<<eBlocks, blk, 0, stream>>>(ACC, dinv, b1, B1, n);
  // layer 2 (no relu)
  k_gemm_wmma<0, 128><<<rowTiles, gblk, 0, stream>>>(B1, W2, nullptr, dinv, B0, ACC);
  k_scatter<<<sBlocks, blk, 0, stream>>>(B0, ACC, srcIdx, dstIdx, e);
  k_epilogue<false><<<eBlocks, blk, 0, stream>>>(ACC, dinv, b2, B1, n);
  // classifier: h = relu(x @ cW1 + cb1) [N,64]; out = h @ cW2 + cb2 [N,8]
  k_gemm_wmma<1, 64><<<rowTiles, gblk, 0, stream>>>(B1, cW1, cb1, nullptr, B0, nullptr);
  k_cls2<<<(n * 8 + 255) / 256, blk, 0, stream>>>(B0, cW2, cb2, out, n);
}